// Classifier_58540404244987
// MI455X (gfx1250) — compile-verified
//
#include <hip/hip_runtime.h>

#define DIM  128
#define NCLS 10
#define MT   4          // m-tiles (of 16 rows) per block -> 64 rows/block
#define ROWS (16 * MT)
#define APAD 132        // padded LDS row stride (floats) -> conflict-free A reads

typedef float v2f_t __attribute__((ext_vector_type(2)));
typedef float v8f_t __attribute__((ext_vector_type(8)));

// ---- CDNA5 async global->LDS copy (16B per lane), tracked by ASYNCcnt ------
__device__ __forceinline__ void async_g2l_b128(uint32_t lds_byte, const float* gp) {
    asm volatile("global_load_async_to_lds_b128 %0, %1, off"
                 :: "v"(lds_byte), "v"(gp) : "memory");
}
__device__ __forceinline__ void wait_async0() {
#if __has_builtin(__builtin_amdgcn_s_wait_asynccnt)
    __builtin_amdgcn_s_wait_asynccnt(0);
#else
    asm volatile("s_wait_asynccnt 0" ::: "memory");
#endif
}

// ---------------------------------------------------------------------------
// int degree histogram over dst
// ---------------------------------------------------------------------------
__global__ __launch_bounds__(256) void degree_kernel(const int* __restrict__ dst,
                                                     int* __restrict__ deg, int E) {
    int e = blockIdx.x * blockDim.x + threadIdx.x;
    if (e < E) atomicAdd(&deg[dst[e]], 1);
}

__global__ __launch_bounds__(256) void invdeg_kernel(const int* __restrict__ deg,
                                                     float* __restrict__ invdeg, int n) {
    int i = blockIdx.x * blockDim.x + threadIdx.x;
    if (i < n) invdeg[i] = 1.0f / fmaxf((float)deg[i], 1.0f);
}

// ---------------------------------------------------------------------------
// single-block chunked exclusive scan of deg -> base   (n ~ 50k, trivial cost)
// ---------------------------------------------------------------------------
__global__ __launch_bounds__(256) void scan_kernel(const int* __restrict__ deg,
                                                   int* __restrict__ base, int n) {
    __shared__ int buf[256];
    __shared__ int carry_s;
    if (threadIdx.x == 0) carry_s = 0;
    __syncthreads();
    for (int start = 0; start < n; start += 256) {
        int i = start + threadIdx.x;
        int v = (i < n) ? deg[i] : 0;
        buf[threadIdx.x] = v;
        __syncthreads();
        int x = v;
        for (int off = 1; off < 256; off <<= 1) {
            int t = (threadIdx.x >= off) ? buf[threadIdx.x - off] : 0;
            __syncthreads();
            x += t;
            buf[threadIdx.x] = x;
            __syncthreads();
        }
        int carry = carry_s;
        if (i < n) base[i] = carry + x - v;  // exclusive
        __syncthreads();
        if (threadIdx.x == 255) carry_s = carry + x;
        __syncthreads();
    }
}

// ---------------------------------------------------------------------------
// CSR fill: csr[base[d] + cursor[d]++] = src[e]
// ---------------------------------------------------------------------------
__global__ __launch_bounds__(256) void fill_kernel(const int* __restrict__ src,
                                                   const int* __restrict__ dst,
                                                   const int* __restrict__ base,
                                                   int* __restrict__ cursor,
                                                   int* __restrict__ csr, int E) {
    int e = blockIdx.x * blockDim.x + threadIdx.x;
    if (e < E) {
        int d = dst[e];
        int pos = base[d] + atomicAdd(&cursor[d], 1);
        csr[pos] = src[e];
    }
}

// ---------------------------------------------------------------------------
// gather: agg[n] = mean of h[neighbors(n)].  Wave per node, lane = 4 columns.
// One edge = 32 lanes x 16B = fully coalesced 512B row read. No atomics.
// ---------------------------------------------------------------------------
__global__ __launch_bounds__(256) void gather_kernel(const float* __restrict__ h,
                                                     const int* __restrict__ csr,
                                                     const int* __restrict__ base,
                                                     const int* __restrict__ deg,
                                                     const float* __restrict__ invdeg,
                                                     float* __restrict__ agg, int n) {
    int node = blockIdx.x * 8 + (threadIdx.x >> 5);
    int lane = threadIdx.x & 31;
    if (node >= n) return;
    int b0 = base[node];
    int cnt = deg[node];
    float4 acc = {0.f, 0.f, 0.f, 0.f};
    const float* hp = h + lane * 4;
    for (int i = 0; i < cnt; ++i) {
        int s = csr[b0 + i];
        const float4 v = *(const float4*)(hp + (size_t)s * DIM);
        acc.x += v.x; acc.y += v.y; acc.z += v.z; acc.w += v.w;
    }
    float idg = invdeg[node];
    float4 o = {acc.x * idg, acc.y * idg, acc.z * idg, acc.w * idg};
    *(float4*)(agg + (size_t)node * DIM + lane * 4) = o;
}

// ---------------------------------------------------------------------------
// SAGE layer: h_out = relu(h @ w_self + agg @ w_neigh + bias)
// Block = 64 node rows. 8 waves = 8 column tiles of 16; each wave keeps MT=4
// f32 16x16 accumulators. A(h,agg) staged once, W staged in two 64-K slabs,
// all via async global->LDS b128; WMMA fed from LDS (no exec divergence).
// ---------------------------------------------------------------------------
__global__ __launch_bounds__(256) void sage_layer_kernel(
    const float* __restrict__ h,       // [N,D]
    const float* __restrict__ agg,     // [N,D] (already mean-scaled)
    const float* __restrict__ w_self,  // [D,D] (k,n)
    const float* __restrict__ w_neigh, // [D,D]
    const float* __restrict__ bias,    // [D]
    float* __restrict__ h_out,         // [N,D]
    int n_nodes) {
    __shared__ float lsA[ROWS * APAD];      // h rows      (33.0 KB)
    __shared__ float lsG[ROWS * APAD];      // agg rows    (33.0 KB)
    __shared__ float lsW[2 * 64 * DIM];     // W slab x2 matrices (64 KB)

    const int row0  = blockIdx.x * ROWS;
    const int wave  = threadIdx.x >> 5;
    const int lane  = threadIdx.x & 31;
    const int col0  = wave * 16;
    const int lhalf = lane >> 4;            // 0: K={0,1}, 1: K={2,3}
    const int lmod  = lane & 15;
    const int col   = col0 + lmod;

    const uint32_t la_base = (uint32_t)(uintptr_t)&lsA[0];
    const uint32_t lg_base = (uint32_t)(uintptr_t)&lsG[0];
    const uint32_t lw_base = (uint32_t)(uintptr_t)&lsW[0];

    // ---- stage A (h + agg), rows clamped so EXEC stays full ----
    for (int i = 0; i < 8; ++i) {
        int f   = threadIdx.x + 256 * i;    // float4 id, 2048 per array
        int row = f >> 5;                   // 32 float4 per row
        int c4  = (f & 31) * 4;
        int g   = row0 + row;
        if (g >= n_nodes) g = n_nodes - 1;
        uint32_t lo = (uint32_t)((row * APAD + c4) * 4);
        async_g2l_b128(la_base + lo, h   + (size_t)g * DIM + c4);
        async_g2l_b128(lg_base + lo, agg + (size_t)g * DIM + c4);
    }

    v8f_t acc[MT];
#pragma unroll
    for (int t = 0; t < MT; ++t) acc[t] = (v8f_t){0.f,0.f,0.f,0.f,0.f,0.f,0.f,0.f};

    for (int slab = 0; slab < 2; ++slab) {
        const int kbase = slab * 64;
        if (slab) __syncthreads();          // prev slab fully consumed
        // ---- stage both weight matrices' 64-K slab ----
        for (int i = 0; i < 16; ++i) {
            int f    = threadIdx.x + 256 * i;   // float4 id, 4096 total
            int m    = f >> 11;                 // 0: w_self, 1: w_neigh
            int r    = f & 2047;
            int krow = r >> 5;
            int c4   = (r & 31) * 4;
            const float* gp = (m ? w_neigh : w_self) + (size_t)(kbase + krow) * DIM + c4;
            async_g2l_b128(lw_base + (uint32_t)((m * 8192 + krow * DIM + c4) * 4), gp);
        }
        wait_async0();
        __syncthreads();

        // ---- WMMA sweep over this slab ----
        for (int kk = 0; kk < 64; kk += 4) {
            const int ka = kbase + kk + 2 * lhalf;  // global K for A
            const int kw = kk + 2 * lhalf;          // slab-local K for W
            v2f_t b_s = {lsW[kw * DIM + col],        lsW[(kw + 1) * DIM + col]};
            v2f_t b_n = {lsW[8192 + kw * DIM + col], lsW[8192 + (kw + 1) * DIM + col]};
#pragma unroll
            for (int t = 0; t < MT; ++t) {
                const int ar = (t * 16 + lmod) * APAD + ka;
                v2f_t a_h = {lsA[ar], lsA[ar + 1]};
                v2f_t a_g = {lsG[ar], lsG[ar + 1]};
                acc[t] = __builtin_amdgcn_wmma_f32_16x16x4_f32(false, a_h, false, b_s,
                                                               (short)0, acc[t], false, false);
                acc[t] = __builtin_amdgcn_wmma_f32_16x16x4_f32(false, a_g, false, b_n,
                                                               (short)0, acc[t], false, false);
            }
        }
    }

    // ---- epilogue: bias + relu; C/D VGPR j -> row (j + 8*lhalf) ----
    const float bv = bias[col];
#pragma unroll
    for (int t = 0; t < MT; ++t) {
#pragma unroll
        for (int j = 0; j < 8; ++j) {
            int orow = row0 + t * 16 + j + 8 * lhalf;
            if (orow < n_nodes) {
                float v = acc[t][j] + bv;
                h_out[(size_t)orow * DIM + col] = v > 0.f ? v : 0.f;
            }
        }
    }
}

// ---------------------------------------------------------------------------
// readout: hg[gid[n]] += h[n];  cnt[gid[n]] += 1
// ---------------------------------------------------------------------------
__global__ __launch_bounds__(256) void readout_kernel(const float* __restrict__ h,
                                                      const int* __restrict__ gid,
                                                      float* __restrict__ hg,
                                                      float* __restrict__ cnt, int n_nodes) {
    int tid = blockIdx.x * blockDim.x + threadIdx.x;
    int n = tid >> 5;
    int c = (tid & 31) * 4;
    if (n >= n_nodes) return;
    int g = gid[n];
    if (c == 0) atomicAdd(&cnt[g], 1.0f);
    const float4 v = *(const float4*)(h + (size_t)n * DIM + c);
    float* p = hg + (size_t)g * DIM + c;
    atomicAdd(p + 0, v.x);
    atomicAdd(p + 1, v.y);
    atomicAdd(p + 2, v.z);
    atomicAdd(p + 3, v.w);
}

// ---------------------------------------------------------------------------
// classifier: out[g,c] = (hg[g]/max(cnt[g],1)) . w_cls[:,c] + b_cls[c]
// ---------------------------------------------------------------------------
__global__ __launch_bounds__(32) void cls_kernel(const float* __restrict__ hg,
                                                 const float* __restrict__ cnt,
                                                 const float* __restrict__ w_cls,
                                                 const float* __restrict__ b_cls,
                                                 float* __restrict__ out) {
    int g = blockIdx.x;
    int cc = threadIdx.x;
    if (cc >= NCLS) return;
    float inv = 1.0f / fmaxf(cnt[g], 1.0f);
    float acc = b_cls[cc];
    const float* hgp = hg + g * DIM;
    for (int k = 0; k < DIM; ++k) acc += hgp[k] * inv * w_cls[k * NCLS + cc];
    out[g * NCLS + cc] = acc;
}

// ---------------------------------------------------------------------------
extern "C" void kernel_launch(void* const* d_in, const int* in_sizes, int n_in,
                              void* d_out, int out_size, void* d_ws, size_t ws_size,
                              hipStream_t stream) {
    const float* h_in    = (const float*)d_in[0];
    const int*   src     = (const int*)d_in[1];
    const int*   dst     = (const int*)d_in[2];
    const int*   gid     = (const int*)d_in[3];
    const float* w_self  = (const float*)d_in[5];  // [L,D,D]
    const float* w_neigh = (const float*)d_in[6];  // [L,D,D]
    const float* b       = (const float*)d_in[7];  // [L,D]
    const float* w_cls   = (const float*)d_in[8];  // [D,C]
    const float* b_cls   = (const float*)d_in[9];  // [C]
    float* out = (float*)d_out;

    const int N = in_sizes[0] / DIM;
    const int E = in_sizes[1];
    const int G = out_size / NCLS;
    const int L = in_sizes[7] / DIM;

    size_t off = 0;
    auto carve = [&](size_t bytes) {
        void* p = (char*)d_ws + off;
        off += (bytes + 255) & ~(size_t)255;
        return p;
    };
    float* hbuf0  = (float*)carve((size_t)N * DIM * sizeof(float));
    float* hbuf1  = (float*)carve((size_t)N * DIM * sizeof(float));
    float* agg    = (float*)carve((size_t)N * DIM * sizeof(float));
    float* invdeg = (float*)carve((size_t)N * sizeof(float));
    int*   degi   = (int*)carve((size_t)N * sizeof(int));
    int*   base   = (int*)carve((size_t)N * sizeof(int));
    int*   cursor = (int*)carve((size_t)N * sizeof(int));
    int*   csr    = (int*)carve((size_t)E * sizeof(int));
    float* hg     = (float*)carve((size_t)G * DIM * sizeof(float));
    float* cnt    = (float*)carve((size_t)G * sizeof(float));

    // 1) degree -> invdeg, CSR build (once per launch)
    hipMemsetAsync(degi, 0, (size_t)N * sizeof(int), stream);
    degree_kernel<<<(E + 255) / 256, 256, 0, stream>>>(dst, degi, E);
    invdeg_kernel<<<(N + 255) / 256, 256, 0, stream>>>(degi, invdeg, N);
    scan_kernel<<<1, 256, 0, stream>>>(degi, base, N);
    hipMemsetAsync(cursor, 0, (size_t)N * sizeof(int), stream);
    fill_kernel<<<(E + 255) / 256, 256, 0, stream>>>(src, dst, base, cursor, csr, E);

    // 2) L SAGE layers: CSR gather (mean) then WMMA layer
    const float* cur = h_in;
    float* bufs[2] = {hbuf0, hbuf1};
    const int gather_blocks = (N + 7) / 8;
    const int gemm_blocks   = (N + ROWS - 1) / ROWS;
    for (int l = 0; l < L; ++l) {
        gather_kernel<<<gather_blocks, 256, 0, stream>>>(cur, csr, base, degi, invdeg, agg, N);
        float* nxt = bufs[l & 1];
        sage_layer_kernel<<<gemm_blocks, 256, 0, stream>>>(
            cur, agg,
            w_self + (size_t)l * DIM * DIM,
            w_neigh + (size_t)l * DIM * DIM,
            b + (size_t)l * DIM,
            nxt, N);
        cur = nxt;
    }

    // 3) readout + classifier
    hipMemsetAsync(hg, 0, (size_t)G * DIM * sizeof(float), stream);
    hipMemsetAsync(cnt, 0, (size_t)G * sizeof(float), stream);
    readout_kernel<<<(N * 32 + 255) / 256, 256, 0, stream>>>(cur, gid, hg, cnt, N);
    cls_kernel<<<G, 32, 0, stream>>>(hg, cnt, w_cls, b_cls, out);
}